// Geoneigh_GeoAtt_37855841747280
// MI455X (gfx1250) — compile-verified
//
#include <hip/hip_runtime.h>
#include <cstdint>

// Problem constants (B=2, N=4096, K=16, H=8, D=64)
#define KN   16
#define HN   8
#define DN   64
#define BN_TOTAL 8192          // B*N blocks
#define SLAB_FLOATS (KN*HN*DN) // 8192 floats = 32KB per (b,n) per tensor

// ---- CDNA5 async global->LDS DMA (ASYNCcnt path) ------------------------
__device__ __forceinline__ void async_ld16(uint32_t lds_byte_off, const void* gaddr) {
  // GLOBAL_LOAD_ASYNC_TO_LDS_B128: VDST = LDS byte address (VGPR), VADDR = 64-bit global addr
  asm volatile("global_load_async_to_lds_b128 %0, %1, off"
               :: "v"(lds_byte_off), "v"(gaddr)
               : "memory");
}
__device__ __forceinline__ void wait_async0() {
  asm volatile("s_wait_asynccnt 0" ::: "memory");
}

__global__ __launch_bounds__(256)
void Geoneigh_GeoAtt_kernel(const float* __restrict__ q,
                            const float* __restrict__ kn,
                            const float* __restrict__ vn,
                            const float* __restrict__ gf,
                            float* __restrict__ out) {
  // 64KB LDS: s_a holds k_neigh slab, later reused for v_neigh; s_b holds geo_f
  __shared__ __align__(16) float2 s_a[SLAB_FLOATS / 2];  // 32KB
  __shared__ __align__(16) float2 s_b[SLAB_FLOATS / 2];  // 32KB

  const int t    = threadIdx.x;        // 0..255
  const int bn   = blockIdx.x;         // 0..8191 == (b*N + n)
  const int h    = t >> 5;             // 0..7
  const int lane = t & 31;             // d-pair index, d = lane*2

  const size_t base3 = (size_t)bn * SLAB_FLOATS;
  const char* kn_g = (const char*)(kn + base3);
  const char* gf_g = (const char*)(gf + base3);
  const char* v_g  = (const char*)(vn + base3);

  // Low 32 bits of a generic LDS pointer == LDS byte offset on gfx1250
  const uint32_t sa = (uint32_t)(uintptr_t)(&s_a[0]);
  const uint32_t sb = (uint32_t)(uintptr_t)(&s_b[0]);

  // Phase 1 DMA: stage k_neigh -> s_a and geo_f -> s_b.
  // 256 threads * 16B * 8 passes = 32KB per tensor; 512B contiguous per wave issue.
#pragma unroll
  for (int p = 0; p < 8; ++p) {
    const uint32_t off = (uint32_t)(p * 4096 + t * 16);
    async_ld16(sa + off, kn_g + off);
    async_ld16(sb + off, gf_g + off);
  }

  // q is tiny (16MB total): direct coalesced float2 load, overlapped with DMA
  const size_t oidx = (size_t)bn * (HN * DN) + h * DN + lane * 2;
  const float2 q2 = *(const float2*)(q + oidx);

  wait_async0();
  __syncthreads();

  // Phase 2: logits + running max (softmax over k, elementwise in d)
  const float scale = 0.125f;  // 64^-0.5
  float w0[KN], w1[KN];
  float m0 = -3.402823466e38f, m1 = -3.402823466e38f;
#pragma unroll
  for (int k = 0; k < KN; ++k) {
    const int idx = k * 256 + h * 32 + lane;   // float2 units; conflict-free ds_load_b64
    const float2 a = s_a[idx];
    const float2 g = s_b[idx];
    const float x0 = q2.x * (a.x + g.x) * scale;
    const float x1 = q2.y * (a.y + g.y) * scale;
    w0[k] = x0; w1[k] = x1;
    m0 = fmaxf(m0, x0); m1 = fmaxf(m1, x1);
  }

  __syncthreads();   // all waves done reading s_a -> safe to overwrite with v_neigh

  // Phase 3 DMA: v_neigh -> s_a, overlapped with the exp/sum VALU work below
#pragma unroll
  for (int p = 0; p < 8; ++p) {
    const uint32_t off = (uint32_t)(p * 4096 + t * 16);
    async_ld16(sa + off, v_g + off);
  }

  float s0 = 0.f, s1 = 0.f;
#pragma unroll
  for (int k = 0; k < KN; ++k) {
    const float e0 = __expf(w0[k] - m0);
    const float e1 = __expf(w1[k] - m1);
    w0[k] = e0; w1[k] = e1;
    s0 += e0; s1 += e1;
  }

  wait_async0();
  __syncthreads();

  // Phase 4: weighted sum over neighbors
  float acc0 = 0.f, acc1 = 0.f;
#pragma unroll
  for (int k = 0; k < KN; ++k) {
    const int idx = k * 256 + h * 32 + lane;
    const float2 v2 = s_a[idx];
    acc0 = fmaf(w0[k], v2.x, acc0);
    acc1 = fmaf(w1[k], v2.y, acc1);
  }

  float2 o;
  o.x = acc0 / s0;
  o.y = acc1 / s1;
  *(float2*)(out + oidx) = o;
}

extern "C" void kernel_launch(void* const* d_in, const int* in_sizes, int n_in,
                              void* d_out, int out_size, void* d_ws, size_t ws_size,
                              hipStream_t stream) {
  // setup_inputs() dict order: q, k_neigh, v_neigh, geo_f
  const float* q  = (const float*)d_in[0];
  const float* kn = (const float*)d_in[1];
  const float* vn = (const float*)d_in[2];
  const float* gf = (const float*)d_in[3];
  float* out = (float*)d_out;

  dim3 grid(BN_TOTAL);   // one block per (b, n)
  dim3 block(256);       // 8 waves (wave32); 64KB LDS -> 5 blocks/WGP resident
  Geoneigh_GeoAtt_kernel<<<grid, block, 0, stream>>>(q, kn, vn, gf, out);
}